// SegmentLabel_54022098649486
// MI455X (gfx1250) — compile-verified
//
#include <hip/hip_runtime.h>

#define NUM_LABELS 150
// min_pixels = max(int(1024*1024*0.2), 10) = 209715  (static in the reference)
#define MIN_PIXELS 209715u

// Native clang vector types: required by __builtin_nontemporal_load/store,
// and they lower to 128-bit global_load_b128 / global_store_b128.
typedef int   vi4 __attribute__((ext_vector_type(4)));
typedef float vf4 __attribute__((ext_vector_type(4)));

// ---------------------------------------------------------------------------
// Kernel 1: LDS-privatized histogram of the 32M-element label map.
// Default (RT) temporal hint on the segment loads deliberately leaves the
// 128MB segment tensor resident in the 192MB L2 -- the gather pass re-reads
// it from L2 instead of HBM.
// ---------------------------------------------------------------------------
__global__ void seg_hist_kernel(const int* __restrict__ seg,
                                unsigned int* __restrict__ counts,
                                int n4, int n) {
    __shared__ unsigned int lhist[NUM_LABELS];
    for (int i = threadIdx.x; i < NUM_LABELS; i += blockDim.x) lhist[i] = 0u;
    __syncthreads();

    const vi4* __restrict__ seg4 = (const vi4*)seg;
    const int stride = gridDim.x * blockDim.x;

    // 128-bit streaming loads (global_load_b128), 4 labels per lane per trip.
#pragma unroll 4
    for (int i = blockIdx.x * blockDim.x + threadIdx.x; i < n4; i += stride) {
        vi4 v = seg4[i];
        atomicAdd(&lhist[v.x], 1u);
        atomicAdd(&lhist[v.y], 1u);
        atomicAdd(&lhist[v.z], 1u);
        atomicAdd(&lhist[v.w], 1u);
    }
    // Scalar tail (n is 32M here, so this normally does nothing).
    for (int i = n4 * 4 + blockIdx.x * blockDim.x + threadIdx.x; i < n;
         i += stride) {
        atomicAdd(&lhist[seg[i]], 1u);
    }
    __syncthreads();

    for (int i = threadIdx.x; i < NUM_LABELS; i += blockDim.x) {
        unsigned int c = lhist[i];
        if (c) atomicAdd(&counts[i], c);
    }
}

// ---------------------------------------------------------------------------
// Kernel 2: build the 150-entry remap LUT (single workgroup, trivial cost).
// guide[l]  = counts[l] >= MIN_PIXELS
// small[l]  = 0 < counts[l] < MIN_PIXELS
// For small labels: first candidate down column l of label_mapping that is a
// guide label; if none, the LAST candidate in the column (reference fallback).
// LUT is stored pre-converted to float (harness output buffer is float*).
// ---------------------------------------------------------------------------
__global__ void build_lut_kernel(const unsigned int* __restrict__ counts,
                                 const int* __restrict__ mapping,
                                 float* __restrict__ lut) {
    __shared__ unsigned char guide[NUM_LABELS];
    const int t = threadIdx.x;

    if (t < NUM_LABELS) {
        guide[t] = (counts[t] >= MIN_PIXELS) ? 1 : 0;
    }
    __syncthreads();

    if (t < NUM_LABELS) {
        const unsigned int c = counts[t];
        const bool small = (c > 0u) && (c < MIN_PIXELS);
        int result = t;  // identity by default
        if (small) {
            int chosen = mapping[(NUM_LABELS - 1) * NUM_LABELS + t];  // fallback: last row
            for (int k = 0; k < NUM_LABELS; ++k) {
                const int cand = mapping[k * NUM_LABELS + t];
                if (guide[cand]) { chosen = cand; break; }           // first guide hit
            }
            result = chosen;
        }
        lut[t] = (float)result;
    }
}

// ---------------------------------------------------------------------------
// Kernel 3: fused gather  out[i] = lut[segment[i]].
// LUT lives in LDS (150 floats).  Hot loop:
//   global_load_b128 th:NT (segment lines are L2-resident from pass 1 and
//   dead after this read)  ->  4x ds_load_b32  ->
//   global_store_b128 th:NT (output stream must NOT evict segment from L2).
// ---------------------------------------------------------------------------
__global__ void lut_gather_kernel(const int* __restrict__ seg,
                                  const float* __restrict__ lut_g,
                                  float* __restrict__ out,
                                  int n4, int n) {
    __shared__ float flut[NUM_LABELS];
    for (int i = threadIdx.x; i < NUM_LABELS; i += blockDim.x)
        flut[i] = lut_g[i];
    __syncthreads();

    const vi4* __restrict__ seg4 = (const vi4*)seg;
    vf4* __restrict__ out4 = (vf4*)out;
    const int stride = gridDim.x * blockDim.x;

#pragma unroll 4
    for (int i = blockIdx.x * blockDim.x + threadIdx.x; i < n4; i += stride) {
        vi4 v = __builtin_nontemporal_load(&seg4[i]);   // last use of this line
        vf4 r;
        r.x = flut[v.x];
        r.y = flut[v.y];
        r.z = flut[v.z];
        r.w = flut[v.w];
        __builtin_nontemporal_store(r, &out4[i]);       // don't pollute L2
    }
    for (int i = n4 * 4 + blockIdx.x * blockDim.x + threadIdx.x; i < n;
         i += stride) {
        float r = flut[__builtin_nontemporal_load(&seg[i])];
        __builtin_nontemporal_store(r, &out[i]);
    }
}

// ---------------------------------------------------------------------------
// Launcher.  d_in[0] = segment (int32, 32*1*1024*1024)
//            d_in[1] = label_mapping (int32, 150*150)
//            d_out   = float, 32*1*1024*1024
// d_ws layout: [0, 600)    counts  (150 x u32)  -- zeroed each launch
//              [1024,1624) lut     (150 x f32)
// ---------------------------------------------------------------------------
extern "C" void kernel_launch(void* const* d_in, const int* in_sizes, int n_in,
                              void* d_out, int out_size, void* d_ws, size_t ws_size,
                              hipStream_t stream) {
    const int* seg = (const int*)d_in[0];
    const int* mapping = (const int*)d_in[1];
    float* out = (float*)d_out;

    const int n = in_sizes[0];
    const int n4 = n / 4;

    unsigned int* counts = (unsigned int*)d_ws;
    float* lut = (float*)((char*)d_ws + 1024);

    // Zero the histogram bins (graph-capture-safe memset node).
    (void)hipMemsetAsync(counts, 0, NUM_LABELS * sizeof(unsigned int), stream);

    // Histogram: 2048 blocks x 256 threads (wave32: 8 waves/block);
    // each thread handles 16 int4 vectors -> deep loadcnt pipelining.
    {
        int blocks = 2048;
        int maxBlocks = (n4 + 255) / 256;
        if (blocks > maxBlocks && maxBlocks > 0) blocks = maxBlocks;
        if (blocks < 1) blocks = 1;
        seg_hist_kernel<<<blocks, 256, 0, stream>>>(seg, counts, n4, n);
    }

    // LUT build: single block, 160 threads (5 waves) covers 150 labels.
    build_lut_kernel<<<1, 160, 0, stream>>>(counts, mapping, lut);

    // Gather: 8192 blocks x 256 threads, 4 int4 vectors per thread.
    {
        int blocks = 8192;
        int maxBlocks = (n4 + 255) / 256;
        if (blocks > maxBlocks && maxBlocks > 0) blocks = maxBlocks;
        if (blocks < 1) blocks = 1;
        lut_gather_kernel<<<blocks, 256, 0, stream>>>(seg, lut, out, n4, n);
    }
}